// GCNConv_4440996184785
// MI455X (gfx1250) — compile-verified
//
#include <hip/hip_runtime.h>
#include <hip/hip_bf16.h>

typedef __attribute__((ext_vector_type(2))) float v2f;
typedef __attribute__((ext_vector_type(8))) float v8f;

#define FEAT_IN   128
#define FEAT_OUT  64
#define LDS_STRIDE 130   // pad 128 -> 130 so B reads (stride 130 floats per n) hit distinct banks

// ---------------------------------------------------------------------------
// Zero a float buffer (deg accumulator).
__global__ void gcn_zero_kernel(float* __restrict__ p, int n) {
    int i = blockIdx.x * blockDim.x + threadIdx.x;
    if (i < n) p[i] = 0.0f;
}

// ---------------------------------------------------------------------------
// deg[row[e]] += 1  over the original edge list (f32 atomics, L2-resident).
__global__ void gcn_degree_kernel(const long long* __restrict__ rows,
                                  float* __restrict__ deg, long long nE) {
    long long e = (long long)blockIdx.x * blockDim.x + threadIdx.x;
    if (e < nE) {
        int r = (int)rows[e];
        __hip_atomic_fetch_add(&deg[r], 1.0f, __ATOMIC_RELAXED,
                               __HIP_MEMORY_SCOPE_AGENT);
    }
}

// ---------------------------------------------------------------------------
// dis[i] = deg[i]^-0.5  (rsqrtf(0) = +inf, matching jnp deg**-0.5). In place.
__global__ void gcn_dis_kernel(float* __restrict__ d, int n) {
    int i = blockIdx.x * blockDim.x + threadIdx.x;
    if (i < n) d[i] = __frsqrt_rn(d[i]);
}

// ---------------------------------------------------------------------------
// h = x @ W via V_WMMA_F32_16X16X4_F32, one wave per 16x64 tile.
// Fused: out = dis^2 * h + bias   (self-loop term + bias init of d_out).
__global__ __launch_bounds__(256)
void gcn_gemm_kernel(const float* __restrict__ x,
                     const float* __restrict__ W,
                     const float* __restrict__ bias,
                     const float* __restrict__ dis,
                     float* __restrict__ h,
                     float* __restrict__ out,
                     int nNodes) {
    __shared__ float Wt[FEAT_OUT * LDS_STRIDE];   // W transposed: Wt[n][k], 33.3 KB

    // Cooperative transpose-load of the 128x64 weight into LDS.
    for (int i = threadIdx.x; i < FEAT_IN * FEAT_OUT; i += blockDim.x) {
        int k = i >> 6;       // row of W
        int n = i & 63;       // col of W
        Wt[n * LDS_STRIDE + k] = W[i];
    }
    __syncthreads();

    const int wave = threadIdx.x >> 5;
    const int lane = threadIdx.x & 31;
    const int tile = blockIdx.x * 8 + wave;      // 16-row tile index (100000/16 = 6250 exact)
    const int row0 = tile * 16;
    if (row0 >= nNodes) return;

    const int m    = lane & 15;                  // A: row-in-tile. B/C/D: n-in-16.
    const int koff = (lane >> 4) * 2;            // lanes 0-15 -> K{0,1}; 16-31 -> K{2,3}

    v8f c0 = {}, c1 = {}, c2 = {}, c3 = {};
    const float* xrow = x + (size_t)(row0 + m) * FEAT_IN;
    const float* wt0  = &Wt[(m +  0) * LDS_STRIDE];
    const float* wt1  = &Wt[(m + 16) * LDS_STRIDE];
    const float* wt2  = &Wt[(m + 32) * LDS_STRIDE];
    const float* wt3  = &Wt[(m + 48) * LDS_STRIDE];

#pragma unroll 8
    for (int k = 0; k < FEAT_IN; k += 4) {
        v2f a  = *(const v2f*)(xrow + k + koff);
        v2f b0 = *(const v2f*)(wt0 + k + koff);
        v2f b1 = *(const v2f*)(wt1 + k + koff);
        v2f b2 = *(const v2f*)(wt2 + k + koff);
        v2f b3 = *(const v2f*)(wt3 + k + koff);
        c0 = __builtin_amdgcn_wmma_f32_16x16x4_f32(false, a, false, b0, (short)0, c0, false, false);
        c1 = __builtin_amdgcn_wmma_f32_16x16x4_f32(false, a, false, b1, (short)0, c1, false, false);
        c2 = __builtin_amdgcn_wmma_f32_16x16x4_f32(false, a, false, b2, (short)0, c2, false, false);
        c3 = __builtin_amdgcn_wmma_f32_16x16x4_f32(false, a, false, b3, (short)0, c3, false, false);
    }

    // C/D layout: VGPR r -> M = r (lanes 0-15) or r+8 (lanes 16-31); N = lane&15.
    const int baseM = row0 + (lane >> 4) * 8;
    float dis2[8];
#pragma unroll
    for (int r = 0; r < 8; ++r) {
        float d = dis[baseM + r];
        dis2[r] = d * d;                          // D^{-1} self-loop weight
    }
    float bz0 = bias[ 0 + m], bz1 = bias[16 + m];
    float bz2 = bias[32 + m], bz3 = bias[48 + m];

#pragma unroll
    for (int r = 0; r < 8; ++r) {
        size_t base = (size_t)(baseM + r) * FEAT_OUT + m;
        h[base +  0] = c0[r];
        h[base + 16] = c1[r];
        h[base + 32] = c2[r];
        h[base + 48] = c3[r];
        out[base +  0] = dis2[r] * c0[r] + bz0;
        out[base + 16] = dis2[r] * c1[r] + bz1;
        out[base + 32] = dis2[r] * c2[r] + bz2;
        out[base + 48] = dis2[r] * c3[r] + bz3;
    }
}

// ---------------------------------------------------------------------------
// Edge scatter: ONE WAVE PER EDGE (wave32). Edge id is made wave-uniform via
// readfirstlane so rows[e]/cols[e]/dis[] become scalar (s_load) traffic.
// Lane j handles features 2j, 2j+1: one b64 gather + two f32 atomics.
__global__ __launch_bounds__(256)
void gcn_scatter_kernel(const long long* __restrict__ rows,
                        const long long* __restrict__ cols,
                        const float* __restrict__ dis,
                        const float* __restrict__ h,
                        float* __restrict__ out,
                        long long nE) {
    const int waveInBlk = __builtin_amdgcn_readfirstlane(threadIdx.x >> 5); // SGPR
    const long long e = (long long)blockIdx.x * 8 + waveInBlk;              // uniform
    if (e >= nE) return;
    const int lane = threadIdx.x & 31;

    const long long r = rows[e];                 // uniform -> s_load_b64
    const long long c = cols[e];                 // uniform -> s_load_b64
    if (r == c) return;                          // w_edge zeroed for self-loops
    const float w = dis[(int)r] * dis[(int)c];   // uniform -> s_load_b32 x2

    const float* hrow = h   + (size_t)c * FEAT_OUT + lane * 2;
    float*       orow = out + (size_t)r * FEAT_OUT + lane * 2;
    v2f hv = *(const v2f*)hrow;                  // coalesced 256B gather per edge
    __hip_atomic_fetch_add(&orow[0], w * hv[0], __ATOMIC_RELAXED,
                           __HIP_MEMORY_SCOPE_AGENT);
    __hip_atomic_fetch_add(&orow[1], w * hv[1], __ATOMIC_RELAXED,
                           __HIP_MEMORY_SCOPE_AGENT);
}

// ---------------------------------------------------------------------------
extern "C" void kernel_launch(void* const* d_in, const int* in_sizes, int n_in,
                              void* d_out, int out_size, void* d_ws, size_t ws_size,
                              hipStream_t stream) {
    const float*     x    = (const float*)d_in[0];
    const long long* edge = (const long long*)d_in[1];   // int64 [2, E]
    const float*     W    = (const float*)d_in[2];
    const float*     bias = (const float*)d_in[3];
    float*           out  = (float*)d_out;

    const int       nNodes = in_sizes[0] / FEAT_IN;      // 100000
    const long long nE     = (long long)in_sizes[1] / 2; // 1,600,000
    const long long* rows = edge;
    const long long* cols = edge + nE;

    // Workspace layout: h [nNodes*64 f32] | deg/dis [nNodes f32]
    float* h   = (float*)d_ws;
    float* dis = h + (size_t)nNodes * FEAT_OUT;

    // 1) deg = 0
    gcn_zero_kernel<<<(nNodes + 255) / 256, 256, 0, stream>>>(dis, nNodes);
    // 2) deg[row] += 1 over edges
    {
        long long blocks = (nE + 255) / 256;
        gcn_degree_kernel<<<(unsigned)blocks, 256, 0, stream>>>(rows, dis, nE);
    }
    // 3) dis = deg^-0.5 (in place)
    gcn_dis_kernel<<<(nNodes + 255) / 256, 256, 0, stream>>>(dis, nNodes);
    // 4) h = x@W (WMMA) fused with out = dis^2*h + bias
    {
        int tiles  = nNodes / 16;                 // 6250
        int blocks = (tiles + 7) / 8;             // 8 waves/block, 1 tile/wave
        gcn_gemm_kernel<<<blocks, 256, 0, stream>>>(x, W, bias, dis, h, out, nNodes);
    }
    // 5) edge scatter-add: one wave (32 lanes) per edge, 8 edges per block
    {
        long long blocks = (nE + 7) / 8;
        gcn_scatter_kernel<<<(unsigned)blocks, 256, 0, stream>>>(rows, cols, dis, h, out, nE);
    }
}